// VectorQuantizerEMA_34608846471813
// MI455X (gfx1250) — compile-verified
//
#include <hip/hip_runtime.h>
#include <hip/hip_bf16.h>
#include <math.h>

// Problem constants (match reference)
#define N_TOK 32768          // 32 * 1024 rows
#define DIM   512            // embedding dim
#define KCB   1024           // codebook size
#define DECAY 0.99f
#define ONE_M_DECAY 0.01f
#define EPS_F 1e-05f

#define ARGMIN_ROWS 32       // rows staged per workgroup (2 waves x 16 rows)

typedef __attribute__((ext_vector_type(2))) float v2f;
typedef __attribute__((ext_vector_type(4))) float v4f;
typedef __attribute__((ext_vector_type(8))) float v8f;
typedef unsigned int __attribute__((ext_vector_type(4))) u32x4;
typedef int  __attribute__((ext_vector_type(4))) i32x4;
typedef int  __attribute__((ext_vector_type(8))) i32x8;

// ---------------- workspace layout (bytes) ----------------
#define OFF_IDX     0          // idx        [N_TOK] int    (128 KB)
#define OFF_CN      131072     // code_norms [KCB]  float   (4 KB)
#define OFF_COUNTS  135168     // counts     [KCB]  float   (4 KB)
#define OFF_CSRAW   139264     // cs_raw     [KCB]  float   (4 KB)
#define OFF_SCAL    143360     // scalars    [4]    float   (n_sum, H, sse, pad)
#define OFF_DW      147456     // dw         [KCB*DIM] float (2 MB)
#define OFF_NEWEMB  2244608    // new_emb    [KCB*DIM] float (2 MB)

// ---------------- kernels ----------------

__global__ __launch_bounds__(256)
void vq_zero(float* __restrict__ counts, float* __restrict__ dw, float* __restrict__ scal) {
    int g = blockIdx.x * 256 + threadIdx.x;
    if (g < KCB * DIM) dw[g] = 0.0f;
    if (g < KCB)       counts[g] = 0.0f;
    if (g < 4)         scal[g] = 0.0f;
}

__global__ __launch_bounds__(256)
void vq_code_norms(const float* __restrict__ emb, float* __restrict__ cn) {
    int k = blockIdx.x * 256 + threadIdx.x;
    if (k >= KCB) return;
    const v4f* e = (const v4f*)(emb + (size_t)k * DIM);
    float s = 0.0f;
    #pragma unroll 4
    for (int d = 0; d < DIM / 4; ++d) {
        v4f v = e[d];
        s += v.x * v.x + v.y * v.y + v.z * v.z + v.w * v.w;
    }
    cn[k] = s;
}

// Fused GEMM + argmin:  idx[n] = argmin_k ( ||e_k||^2 - 2 * x_n . e_k )
// Workgroup = 64 threads (2 waves), 32 rows. The 32x512 f32 A-tile (64 KB) is
// staged into LDS ONCE via the Tensor Data Mover (tensor_load_to_lds, waited
// with s_wait_tensorcnt), then re-read 16x from LDS instead of 16x from L2.
// Each wave owns a 16-row C tile; K is looped in chunks of 64 codes so every
// A fragment (ds_load_b64) feeds 4 V_WMMA_F32_16X16X4_F32 issues.
__global__ __launch_bounds__(64)
void vq_argmin_wmma(const float* __restrict__ flat,
                    const float* __restrict__ emb,
                    const float* __restrict__ cnorm,
                    int* __restrict__ idx_out) {
    __shared__ float tileA[ARGMIN_ROWS * DIM];   // 64 KB

    const int warp = threadIdx.x >> 5;           // 0..1
    const int lane = threadIdx.x & 31;
    const int half = lane >> 4;                  // 0: K={0,1}, C rows 0-7 ; 1: K={2,3}, C rows 8-15
    const int l16  = lane & 15;
    const int block_row_base = blockIdx.x * ARGMIN_ROWS;
    const int row_base = block_row_base + warp * 16;

    // ---- TDM: DMA the 2-D tile [ARGMIN_ROWS x DIM] f32 from global into LDS ----
    if (threadIdx.x < 32) {   // one wave issues; EXEC is ignored by TDM
        unsigned long long gaddr =
            (unsigned long long)(uintptr_t)(flat + (size_t)block_row_base * DIM);
        unsigned int lds_base = (unsigned int)(uintptr_t)&tileA[0];  // flat LDS addr[31:0] == LDS offset

        u32x4 g0;
        g0.x = 1u;                                   // count=1 (valid user descriptor)
        g0.y = lds_base;                             // lds_addr
        g0.z = (unsigned int)gaddr;                  // global_addr[31:0]
        g0.w = (unsigned int)((gaddr >> 32) & 0x01FFFFFFu) | 0x80000000u; // global_addr[56:32] | type=2

        i32x8 g1;
        g1[0] = (int)(2u << 16);                     // data_size=2 (4 bytes); no mask/pad/iterate
        g1[1] = (int)((unsigned)DIM << 16);          // tensor_dim0[15:0]=512 (bits 63:48)
        g1[2] = (int)((unsigned)ARGMIN_ROWS << 16);  // tensor_dim0[31:16]=0 | tensor_dim1[15:0]=32
        g1[3] = (int)((unsigned)DIM << 16);          // tensor_dim1[31:16]=0 | tile_dim0=512
        g1[4] = ARGMIN_ROWS;                         // tile_dim1=32, tile_dim2=0
        g1[5] = DIM;                                 // tensor_dim0_stride[31:0]=512
        g1[6] = 0;                                   // stride0[47:32]=0, stride1[15:0]=0
        g1[7] = 0;

        i32x4 gz = {0, 0, 0, 0};                     // groups 2/3 unused (2-D tile)
#if __clang_major__ >= 23
        i32x8 gz8 = {0, 0, 0, 0, 0, 0, 0, 0};
        __builtin_amdgcn_tensor_load_to_lds(g0, g1, gz, gz, gz8, 0);
#else
        __builtin_amdgcn_tensor_load_to_lds(g0, g1, gz, gz, 0);
#endif
        __builtin_amdgcn_s_wait_tensorcnt(0);
    }
    __syncthreads();

    // A fragment base in LDS: A[M = l16][K = kk + 2*half .. +1]  (16x4 f32 layout)
    const float* aL = &tileA[(size_t)(warp * 16 + l16) * DIM + (half << 1)];

    float best_val[8];
    int   best_idx[8];
    #pragma unroll
    for (int r = 0; r < 8; ++r) { best_val[r] = 3.402823466e38f; best_idx[r] = 0; }

    for (int nb = 0; nb < KCB; nb += 64) {
        // B fragment: B[K][N] = E[N][K] -> lane holds code (nb + 16t + l16), K = kk + 2*half
        const float* bptr = emb + (size_t)(nb + l16) * DIM + (half << 1);
        v8f acc[4] = {};
        #pragma unroll 4
        for (int kk = 0; kk < DIM; kk += 4) {
            v2f a  = *(const v2f*)(aL + kk);                 // ds_load_b64
            v2f b0 = *(const v2f*)(bptr + kk);
            v2f b1 = *(const v2f*)(bptr + 16 * DIM + kk);
            v2f b2 = *(const v2f*)(bptr + 32 * DIM + kk);
            v2f b3 = *(const v2f*)(bptr + 48 * DIM + kk);
            acc[0] = __builtin_amdgcn_wmma_f32_16x16x4_f32(false, a, false, b0, (short)0, acc[0], false, false);
            acc[1] = __builtin_amdgcn_wmma_f32_16x16x4_f32(false, a, false, b1, (short)0, acc[1], false, false);
            acc[2] = __builtin_amdgcn_wmma_f32_16x16x4_f32(false, a, false, b2, (short)0, acc[2], false, false);
            acc[3] = __builtin_amdgcn_wmma_f32_16x16x4_f32(false, a, false, b3, (short)0, acc[3], false, false);
        }
        // Fold 4 code-tiles into the running argmin. Ascending scan + strict '<'
        // keeps the first (lowest-index) minimum, matching jnp.argmin.
        #pragma unroll
        for (int t = 0; t < 4; ++t) {
            int code = nb + 16 * t + l16;
            float cn = cnorm[code];
            #pragma unroll
            for (int r = 0; r < 8; ++r) {
                float dist = cn - 2.0f * acc[t][r];
                if (dist < best_val[r]) { best_val[r] = dist; best_idx[r] = code; }
            }
        }
    }

    // C layout: VGPR r -> (M=r, lanes 0-15) and (M=r+8, lanes 16-31), N = l16.
    // Min-reduce across the 16 lanes of each half (xor masks 1..8 stay in-half).
    #pragma unroll
    for (int r = 0; r < 8; ++r) {
        float v = best_val[r]; int bi = best_idx[r];
        #pragma unroll
        for (int m = 8; m >= 1; m >>= 1) {
            float ov = __shfl_xor(v, m, 32);
            int   oi = __shfl_xor(bi, m, 32);
            if (ov < v || (ov == v && oi < bi)) { v = ov; bi = oi; }
        }
        if (l16 == 0) idx_out[row_base + r + 8 * half] = bi;
    }
}

// counts[k] += 1 ; dw[k][:] += flat[n][:]   (one wave per row)
__global__ __launch_bounds__(256)
void vq_scatter(const float* __restrict__ flat, const int* __restrict__ idx,
                float* __restrict__ counts, float* __restrict__ dw) {
    int gwave = (blockIdx.x * 256 + threadIdx.x) >> 5;
    int lane  = threadIdx.x & 31;
    if (gwave >= N_TOK) return;
    int k = idx[gwave];
    if (lane == 0) atomicAdd(&counts[k], 1.0f);
    const float* src = flat + (size_t)gwave * DIM;
    float*       dst = dw + (size_t)k * DIM;
    #pragma unroll
    for (int d = lane; d < DIM; d += 32) atomicAdd(&dst[d], src[d]);
}

// raw EMA cluster size + global sum + entropy accumulation
__global__ __launch_bounds__(256)
void vq_ema_a(const float* __restrict__ ema_cs, const float* __restrict__ counts,
              float* __restrict__ cs_raw, float* __restrict__ scal) {
    int k = blockIdx.x * 256 + threadIdx.x;
    if (k >= KCB) return;
    float raw = ema_cs[k] * DECAY + ONE_M_DECAY * counts[k];
    cs_raw[k] = raw;
    atomicAdd(&scal[0], raw);
    float p = counts[k] * (1.0f / (float)N_TOK);
    atomicAdd(&scal[1], -p * logf(p + 1e-10f));
}

// Laplace smoothing + EMA weights + new embedding
__global__ __launch_bounds__(256)
void vq_ema_b(const float* __restrict__ ema_w, const float* __restrict__ dw,
              const float* __restrict__ cs_raw, const float* __restrict__ scal,
              float* __restrict__ new_emb) {
    int g = blockIdx.x * 256 + threadIdx.x;   // over KCB*DIM
    int k = g >> 9;                            // / DIM
    float n = scal[0];
    float cs = (cs_raw[k] + EPS_F) / (n + (float)KCB * EPS_F) * n;
    float w = ema_w[g] * DECAY + ONE_M_DECAY * dw[g];
    new_emb[g] = w / cs;
}

// quantized[n][d] = new_emb[idx[n]][d]; fused SSE reduction for the loss
__global__ __launch_bounds__(256)
void vq_quantize(const float* __restrict__ inputs, const int* __restrict__ idx,
                 const float* __restrict__ new_emb, float* __restrict__ out_q,
                 float* __restrict__ scal) {
    int g = blockIdx.x * 256 + threadIdx.x;   // over N_TOK*DIM
    int n = g >> 9;
    int d = g & (DIM - 1);
    int k = idx[n];
    float q = new_emb[(size_t)k * DIM + d];
    out_q[g] = q;   // straight-through output == quantized value numerically
    float diff = q - inputs[g];
    float e = diff * diff;
    #pragma unroll
    for (int m = 16; m >= 1; m >>= 1) e += __shfl_xor(e, m, 32);
    if ((threadIdx.x & 31) == 0) atomicAdd(&scal[2], e);
}

// one-hot encodings, 2 codes per thread via b64 stores (region is 8B aligned)
__global__ __launch_bounds__(256)
void vq_encodings(const int* __restrict__ idx, v2f* __restrict__ enc) {
    int g = blockIdx.x * 256 + threadIdx.x;    // over N_TOK*KCB/2
    int n  = g >> 9;                           // KCB/2 = 512 pairs per row
    int c2 = g & 511;
    int k = idx[n];
    v2f v;
    v.x = (k == 2 * c2)     ? 1.0f : 0.0f;
    v.y = (k == 2 * c2 + 1) ? 1.0f : 0.0f;
    enc[g] = v;
}

__global__ void vq_finalize(const float* __restrict__ scal, float* __restrict__ out) {
    out[0] = 0.25f * scal[2] / (float)(N_TOK * DIM);        // loss
    out[1 + (size_t)N_TOK * DIM] = expf(scal[1]);           // perplexity
}

// ---------------- launch ----------------
extern "C" void kernel_launch(void* const* d_in, const int* in_sizes, int n_in,
                              void* d_out, int out_size, void* d_ws, size_t ws_size,
                              hipStream_t stream) {
    const float* inputs  = (const float*)d_in[0];   // [N_TOK, DIM]
    const float* emb_w   = (const float*)d_in[1];   // [KCB, DIM]
    const float* ema_cs  = (const float*)d_in[2];   // [KCB]
    const float* ema_w   = (const float*)d_in[3];   // [KCB, DIM]

    char* ws = (char*)d_ws;
    int*   idx      = (int*)  (ws + OFF_IDX);
    float* cnorm    = (float*)(ws + OFF_CN);
    float* counts   = (float*)(ws + OFF_COUNTS);
    float* cs_raw   = (float*)(ws + OFF_CSRAW);
    float* scal     = (float*)(ws + OFF_SCAL);
    float* dw       = (float*)(ws + OFF_DW);
    float* new_emb  = (float*)(ws + OFF_NEWEMB);

    float* out = (float*)d_out;
    float* out_q   = out + 1;                                // quantized [N_TOK*DIM]
    v2f*   out_enc = (v2f*)(out + 2 + (size_t)N_TOK * DIM);  // encodings [N_TOK*KCB]

    // 1) zero accumulators
    vq_zero<<<(KCB * DIM + 255) / 256, 256, 0, stream>>>(counts, dw, scal);
    // 2) codebook norms
    vq_code_norms<<<KCB / 256, 256, 0, stream>>>(emb_w, cnorm);
    // 3) fused fp32-WMMA GEMM + argmin, A-tile staged by TDM into LDS
    vq_argmin_wmma<<<N_TOK / ARGMIN_ROWS, 64, 0, stream>>>(inputs, emb_w, cnorm, idx);
    // 4) counts + dw segment-sum
    vq_scatter<<<N_TOK * 32 / 256, 256, 0, stream>>>(inputs, idx, counts, dw);
    // 5) EMA phase A (cluster sizes, n, entropy)
    vq_ema_a<<<KCB / 256, 256, 0, stream>>>(ema_cs, counts, cs_raw, scal);
    // 6) EMA phase B (Laplace smoothing, new embedding)
    vq_ema_b<<<KCB * DIM / 256, 256, 0, stream>>>(ema_w, dw, cs_raw, scal, new_emb);
    // 7) quantized gather + SSE
    vq_quantize<<<N_TOK * DIM / 256, 256, 0, stream>>>(inputs, idx, new_emb, out_q, scal);
    // 8) one-hot encodings
    vq_encodings<<<(N_TOK * (KCB / 2)) / 256, 256, 0, stream>>>(idx, out_enc);
    // 9) scalars
    vq_finalize<<<1, 1, 0, stream>>>(scal, out);
}